// FafePredict_89704686944767
// MI455X (gfx1250) — compile-verified
//
#include <hip/hip_runtime.h>
#include <hip/hip_bf16.h>

// ---------------------------------------------------------------------------
// FAFE predict post-processing for MI455X (gfx1250).
// Pure streaming kernel: ~262 MB read + ~229 MB written => ~21 us at 23.3 TB/s.
// Strategy: wave32-coalesced B64/B128 traffic, detection channels staged via
// the CDNA5 async global->LDS engine (ASYNCcnt), everything else nontemporal.
// ---------------------------------------------------------------------------

#define AS1 __attribute__((address_space(1)))
#define AS3 __attribute__((address_space(3)))

typedef float v2f __attribute__((ext_vector_type(2)));
typedef float v4f __attribute__((ext_vector_type(4)));
typedef int   v2i __attribute__((ext_vector_type(2)));

namespace {
constexpr int kNB = 8, kNT = 5, kNA = 2, kNDET = 2, kNREG = 6;
constexpr int kNH = 320, kNW = 320, kNHW = kNH * kNW;      // 102400
constexpr float kStep = 0.2f;                              // 64/320 for both axes
constexpr float kConf = 0.7f;
}  // namespace

// ---- CDNA5 async global->LDS copy (tracked by ASYNCcnt) -------------------
// Builtin signature (from clang diagnostic): first param is a pointer to
// int __vector(2) in the __device__ (AS1/global) address space; LDS side is
// the matching AS3 pointer.
__device__ __forceinline__ void async_lds_b64(const float* g, v2f* l) {
#if __has_builtin(__builtin_amdgcn_global_load_async_to_lds_b64)
  __builtin_amdgcn_global_load_async_to_lds_b64(
      (AS1 v2i*)g, (AS3 v2i*)l, /*offset=*/0, /*cpol=*/0);
#else
  unsigned loff = (unsigned)(unsigned long long)(AS3 char*)(void*)l;
  asm volatile("global_load_async_to_lds_b64 %0, %1, off"
               :: "v"(loff), "v"(g) : "memory");
#endif
}

__device__ __forceinline__ void wait_async0() {
#if __has_builtin(__builtin_amdgcn_s_wait_asynccnt)
  __builtin_amdgcn_s_wait_asynccnt(0);
#else
  asm volatile("s_wait_asynccnt 0x0" ::: "memory");
#endif
}

__device__ __forceinline__ float sigmoidf_(float x) {
  return 1.0f / (1.0f + __expf(-x));
}

__global__ __launch_bounds__(256) void fafe_predict_kernel(
    const float* __restrict__ det, const float* __restrict__ reg,
    float* __restrict__ boxes, float* __restrict__ maskout) {
  // 4 detection channels * 256 lanes * 8B = 8 KB LDS
  __shared__ v2f sdet[kNA * kNDET][256];

  const int tid = threadIdx.x;
  const int bt  = blockIdx.y;            // bt = b*kNT + t
  const int b   = bt / kNT;
  const int t   = bt - b * kNT;
  const int hw0 = (blockIdx.x * 256 + tid) * 2;   // even; pair never wraps a row

  const float* detb = det + (size_t)b * (kNA * kNDET * kNT) * kNHW;
  const float* regb = reg + (size_t)b * (kNA * kNREG * kNT) * kNHW;

  // Stage detection channels through the async DMA path (in flight while the
  // direct regression loads below are issued).
#pragma unroll
  for (int a = 0; a < kNA; ++a)
#pragma unroll
    for (int d = 0; d < kNDET; ++d) {
      const int ch = (a * kNDET + d) * kNT + t;   // reshape(nB,NA,NDET,NT,..)
      async_lds_b64(detb + (size_t)ch * kNHW + hw0, &sdet[a * kNDET + d][tid]);
    }

  // Regression channels: single-use stream -> nontemporal B64 loads.
  v2f r[kNA][kNREG];
#pragma unroll
  for (int a = 0; a < kNA; ++a)
#pragma unroll
    for (int j = 0; j < kNREG; ++j) {
      const int ch = (a * kNREG + j) * kNT + t;   // reshape(nB,NA,NREG,NT,..)
      r[a][j] = __builtin_nontemporal_load(
          (const v2f*)(regb + (size_t)ch * kNHW + hw0));
    }

  wait_async0();
  __syncthreads();

  const v2f d00 = sdet[0][tid], d01 = sdet[1][tid];   // anchor0: class0/1
  const v2f d10 = sdet[2][tid], d11 = sdet[3][tid];   // anchor1: class0/1

  // softmax over 2 classes -> P(class1) = sigmoid(d1 - d0)
  float p[kNA][2];
#pragma unroll
  for (int e = 0; e < 2; ++e) {
    p[0][e] = 1.0f / (1.0f + __expf(d00[e] - d01[e]));
    p[1][e] = 1.0f / (1.0f + __expf(d10[e] - d11[e]));
  }

  // torch repeat+view quirk: both grids depend only on the column w.
  const float col0 = (float)(hw0 % kNW);
  const float anchorL[2] = {3.9f, 1.0f};
  const float anchorW[2] = {1.6f, 0.6f};

#pragma unroll
  for (int a = 0; a < kNA; ++a) {
    float o12[12] __attribute__((aligned(16)));
    v2f m;
#pragma unroll
    for (int e = 0; e < 2; ++e) {
      const float col = col0 + (float)e;
      const float x   = sigmoidf_(r[a][0][e]) * kStep + col * kStep;
      const float y   = sigmoidf_(r[a][1][e]) * kStep + (col - 160.0f) * kStep;
      const float len = __expf(r[a][3][e]) * anchorL[a];   // l from reg[...,3]
      const float wid = __expf(r[a][2][e]) * anchorW[a];   // w from reg[...,2]
      const float ang = atan2f(r[a][5][e], r[a][4][e]);
      o12[e * 6 + 0] = x;
      o12[e * 6 + 1] = y;
      o12[e * 6 + 2] = len;
      o12[e * 6 + 3] = wid;
      o12[e * 6 + 4] = ang;
      o12[e * 6 + 5] = p[a][e];
      const int best = (p[0][e] >= p[1][e]) ? 0 : 1;       // argmax, tie -> 0
      m[e] = (p[a][e] > kConf && a == best) ? 1.0f : 0.0f;
    }
    // boxes_scores layout (nB,nT,nA,nH,nW,6): 12 contiguous floats per thread,
    // 16B-aligned because hw0 is even (48B groups).
    float* ob = boxes + ((size_t)(bt * kNA + a) * kNHW + hw0) * 6;
    __builtin_nontemporal_store(*(const v4f*)&o12[0], (v4f*)(ob + 0));
    __builtin_nontemporal_store(*(const v4f*)&o12[4], (v4f*)(ob + 4));
    __builtin_nontemporal_store(*(const v4f*)&o12[8], (v4f*)(ob + 8));
    __builtin_nontemporal_store(
        m, (v2f*)(maskout + (size_t)(bt * kNA + a) * kNHW + hw0));
  }
}

extern "C" void kernel_launch(void* const* d_in, const int* in_sizes, int n_in,
                              void* d_out, int out_size, void* d_ws, size_t ws_size,
                              hipStream_t stream) {
  const float* det = (const float*)d_in[0];   // (8, 20, 320, 320) f32
  const float* reg = (const float*)d_in[1];   // (8, 60, 320, 320) f32
  float* boxes = (float*)d_out;                                     // 49,152,000 f32
  float* maskout = boxes + (size_t)kNB * kNT * kNA * kNHW * 6;      //  8,192,000 f32

  dim3 grid(kNHW / 512, kNB * kNT);   // (200, 40) blocks, 256 threads (8 wave32)
  fafe_predict_kernel<<<grid, dim3(256), 0, stream>>>(det, reg, boxes, maskout);
  (void)in_sizes; (void)n_in; (void)out_size; (void)d_ws; (void)ws_size;
}